// GraphConv_27874337751119
// MI455X (gfx1250) — compile-verified
//
#include <hip/hip_runtime.h>

typedef float v2f __attribute__((ext_vector_type(2)));
typedef float v8f __attribute__((ext_vector_type(8)));

#define F 128  // F_IN == F_OUT == 128 per reference

// ---------------------------------------------------------------------------
// out[n, f] = bias[f]   (accumulator base, so scatter can atomically add)
// ---------------------------------------------------------------------------
__global__ void gc_init_bias(float* __restrict__ out,
                             const float* __restrict__ bias, int total) {
  int i = blockIdx.x * blockDim.x + threadIdx.x;
  if (i < total) out[i] = bias[i & (F - 1)];
}

// ---------------------------------------------------------------------------
// ReLU in place
// ---------------------------------------------------------------------------
__global__ void gc_relu(float* __restrict__ out, int total) {
  int i = blockIdx.x * blockDim.x + threadIdx.x;
  if (i < total) out[i] = fmaxf(out[i], 0.0f);
}

// ---------------------------------------------------------------------------
// S[n, :] = X[n, :] @ W   via V_WMMA_F32_16X16X4_F32 (fp32 in/out, wave32)
// Block: 256 threads = 8 waves. Block b covers rows [16b, 16b+16);
// wave w covers columns [16w, 16w+16). K-loop: 128/4 = 32 WMMA ops.
//
// Fragment layouts (CDNA5 ISA §7.12.2):
//   A 16x4 f32, 2 VGPRs: lanes 0-15 -> K = k0,k0+1 of row (lane&15)
//                        lanes 16-31 -> K = k0+2,k0+3
//   B 4x16 f32, 2 VGPRs: same pattern with lane -> column
//   C/D 16x16 f32, 8 VGPRs: VGPR i, lanes 0-15 -> M=i, lanes 16-31 -> M=i+8,
//                           N = lane&15
// ---------------------------------------------------------------------------
__global__ void gc_gemm_wmma_f32(const float* __restrict__ X,
                                 const float* __restrict__ W,
                                 float* __restrict__ S, int n_nodes) {
  const int lane = threadIdx.x & 31;
  const int wave = threadIdx.x >> 5;
  const int row_base = blockIdx.x * 16;
  const int rr   = lane & 15;   // A-row within tile / B-col within tile
  const int half = lane >> 4;   // selects K sub-pair
  const int col  = wave * 16 + rr;

  // Branchless clamp so tail tiles read in-bounds (EXEC stays all-ones for WMMA)
  int arow = row_base + rr;
  if (arow >= n_nodes) arow = n_nodes - 1;
  const float* xrow = X + (size_t)arow * F;

  v8f acc = {0.f, 0.f, 0.f, 0.f, 0.f, 0.f, 0.f, 0.f};
#pragma unroll 4
  for (int k0 = 0; k0 < F; k0 += 4) {
    const int kk = k0 + half * 2;
    v2f a = *(const v2f*)(xrow + kk);          // 8B-aligned: kk even, row 512B
    v2f b;
    b.x = W[(size_t)kk * F + col];
    b.y = W[(size_t)(kk + 1) * F + col];
    // (neg_a, A, neg_b, B, c_mod, C, reuse_a, reuse_b)
    acc = __builtin_amdgcn_wmma_f32_16x16x4_f32(false, a, false, b, (short)0,
                                                acc, false, false);
  }

  // Epilogue: one *uniform* bounds branch. Full tiles (the only case when
  // n_nodes % 16 == 0, as here) take 8 straight global_store_b32 with
  // immediate row offsets; only a genuine tail tile pays per-lane guards.
  if (row_base + 16 <= n_nodes) {
    float* srow = S + (size_t)(row_base + half * 8) * F + col;
#pragma unroll
    for (int i = 0; i < 8; ++i) srow[(size_t)i * F] = acc[i];
  } else {
#pragma unroll
    for (int i = 0; i < 8; ++i) {
      const int row = row_base + i + half * 8;
      if (row < n_nodes) S[(size_t)row * F + col] = acc[i];
    }
  }
}

// ---------------------------------------------------------------------------
// out[rows[e], :] += vals[e] * sup[cols[e], :]
// One wave32 per edge; each lane handles 4 consecutive floats (float4 gather,
// 4 fp32 global atomic adds). sup + out are L2-resident (51 MB << 192 MB L2).
// Edge index `e` is forced wave-uniform via readfirstlane so rows/cols/vals
// loads can scalarize (SMEM path) instead of 32 identical vector loads.
// ---------------------------------------------------------------------------
__global__ void gc_spmm_scatter(const int* __restrict__ rows,
                                const int* __restrict__ cols,
                                const float* __restrict__ vals,
                                const float* __restrict__ sup,
                                float* __restrict__ out, int n_edges) {
  const int wave_in_block = __builtin_amdgcn_readfirstlane(threadIdx.x >> 5);
  const int e = blockIdx.x * 8 + wave_in_block;  // 256 threads = 8 wave32s
  if (e >= n_edges) return;
  const int lane = threadIdx.x & 31;

  const int r = rows[e];
  const int c = cols[e];
  const float v = vals[e];

  const float4 g = *(const float4*)(sup + (size_t)c * F + lane * 4);
  float* o = out + (size_t)r * F + lane * 4;
  __hip_atomic_fetch_add(o + 0, v * g.x, __ATOMIC_RELAXED, __HIP_MEMORY_SCOPE_AGENT);
  __hip_atomic_fetch_add(o + 1, v * g.y, __ATOMIC_RELAXED, __HIP_MEMORY_SCOPE_AGENT);
  __hip_atomic_fetch_add(o + 2, v * g.z, __ATOMIC_RELAXED, __HIP_MEMORY_SCOPE_AGENT);
  __hip_atomic_fetch_add(o + 3, v * g.w, __ATOMIC_RELAXED, __HIP_MEMORY_SCOPE_AGENT);
}

// ---------------------------------------------------------------------------
extern "C" void kernel_launch(void* const* d_in, const int* in_sizes, int n_in,
                              void* d_out, int out_size, void* d_ws, size_t ws_size,
                              hipStream_t stream) {
  const float* x     = (const float*)d_in[0];   // [1, N, 128] fp32
  const int*   rows  = (const int*)d_in[1];     // [S, E] int32
  const int*   cols  = (const int*)d_in[2];     // [S, E] int32
  const float* vals  = (const float*)d_in[3];   // [S, E] fp32
  const float* kerns = (const float*)d_in[4];   // [S, 128, 128] fp32
  const float* bias  = (const float*)d_in[5];   // [128] fp32
  float* out = (float*)d_out;                   // [1, N, 128] fp32
  float* sup = (float*)d_ws;                    // N*128 fp32 scratch (25.6 MB)

  const int n_sup   = in_sizes[4] / (F * F);
  const int n_edges = (n_sup > 0) ? (in_sizes[1] / n_sup) : 0;
  const int n_nodes = in_sizes[0] / F;
  const int total   = n_nodes * F;

  gc_init_bias<<<(total + 255) / 256, 256, 0, stream>>>(out, bias, total);

  const int n_tiles     = (n_nodes + 15) / 16;
  const int spmm_blocks = (n_edges + 7) / 8;    // 8 edges (waves) per block
  for (int k = 0; k < n_sup; ++k) {
    gc_gemm_wmma_f32<<<n_tiles, 256, 0, stream>>>(
        x, kerns + (size_t)k * F * F, sup, n_nodes);
    gc_spmm_scatter<<<spmm_blocks, 256, 0, stream>>>(
        rows + (size_t)k * n_edges, cols + (size_t)k * n_edges,
        vals + (size_t)k * n_edges, sup, out, n_edges);
  }

  gc_relu<<<(total + 255) / 256, 256, 0, stream>>>(out, total);
}